// SliceSectorLoss_78271484002324
// MI455X (gfx1250) — compile-verified
//
#include <hip/hip_runtime.h>
#include <math.h>

typedef __attribute__((ext_vector_type(2))) float v2f;
typedef __attribute__((ext_vector_type(8))) float v8f;

#define DIMS      256
#define NSEC      10
#define SEC_PAD   16
#define CHUNK     512          // source rows per block in the WMMA segment-sum pass

// ---------------------------------------------------------------------------
// ws layout (floats):  [0 .. 4095] sums[16][256]
//                      [4096..4111] counts[16]
//                      [4112]       distance-sum accumulator
// ---------------------------------------------------------------------------

__global__ void zero_ws_kernel(float* ws, int n) {
  int i = blockIdx.x * blockDim.x + threadIdx.x;
  if (i < n) ws[i] = 0.0f;
}

// Segment-sum via V_WMMA_F32_16X16X4_F32:  C[16x16] += onehot(A:16x4) * emb(B:4x16)
__global__ __launch_bounds__(512) void segsum_wmma_kernel(
    const float* __restrict__ emb, const int* __restrict__ sec,
    float* __restrict__ sums, int nrows) {
  __shared__ int ssec[CHUNK];

  const int tid     = threadIdx.x;
  const int lane    = tid & 31;
  const int wave    = tid >> 5;          // 0..15, one 16-col tile each
  const int colBase = wave << 4;
  const int m       = lane & 15;         // sector row (A) / column (B,C)
  const int koff    = (lane >> 4) << 1;  // 0 for lanes 0-15, 2 for lanes 16-31
  const long long rowBase = (long long)blockIdx.x * CHUNK;
  const long long lastRow = (long long)nrows - 1;

  {
    long long r = rowBase + tid;
    ssec[tid] = (r < nrows) ? sec[r] : -1;   // -1 => one-hot column all zero
  }
  __syncthreads();

  v8f acc = {};
#pragma unroll 4
  for (int k = 0; k < CHUNK; k += 4) {
    const int s0 = ssec[k + koff];
    const int s1 = ssec[k + koff + 1];
    v2f a, b;
    a[0] = (s0 == m) ? 1.0f : 0.0f;        // A: lanes<16 K=0,1 ; lanes>=16 K=2,3
    a[1] = (s1 == m) ? 1.0f : 0.0f;
    long long r0 = rowBase + k + koff;     // B: same K split, clamp keeps loads in-bounds
    long long r1 = r0 + 1;
    if (r0 > lastRow) r0 = lastRow;
    if (r1 > lastRow) r1 = lastRow;
    b[0] = emb[r0 * DIMS + colBase + m];
    b[1] = emb[r1 * DIMS + colBase + m];
    acc = __builtin_amdgcn_wmma_f32_16x16x4_f32(
        false, a, false, b, (short)0, acc, false, false);
  }

  // C layout: VGPR r -> sector = r + (lane>=16 ? 8 : 0), col = colBase + (lane&15)
  const int secOff = (lane >> 4) << 3;
#pragma unroll
  for (int r = 0; r < 8; ++r) {
    const int s = r + secOff;
    if (s < NSEC) atomicAdd(&sums[s * DIMS + colBase + m], acc[r]);
  }
}

__global__ void count_kernel(const int* __restrict__ sec,
                             float* __restrict__ counts, int n) {
  __shared__ int h[SEC_PAD];
  if (threadIdx.x < SEC_PAD) h[threadIdx.x] = 0;
  __syncthreads();
  const long long stride = (long long)gridDim.x * blockDim.x;
  for (long long i = (long long)blockIdx.x * blockDim.x + threadIdx.x; i < n; i += stride)
    atomicAdd(&h[sec[i] & 15], 1);
  __syncthreads();
  if (threadIdx.x < SEC_PAD && h[threadIdx.x] != 0)
    atomicAdd(&counts[threadIdx.x], (float)h[threadIdx.x]);
}

// One target per wave32: 8 cols/lane, shfl_xor tree reduce, sqrt, per-wave atomic.
__global__ __launch_bounds__(256) void dist_kernel(
    const float* __restrict__ tgt, const float* __restrict__ slice,
    const float* __restrict__ sums, const float* __restrict__ counts,
    float* __restrict__ acc, int ntgt) {
  __shared__ float centers[NSEC * DIMS];
  for (int i = threadIdx.x; i < NSEC * DIMS; i += blockDim.x) {
    const int s = i >> 8;
    centers[i] = sums[i] / counts[s];
  }
  __syncthreads();

  const int lane   = threadIdx.x & 31;
  const int waveId = (blockIdx.x * blockDim.x + threadIdx.x) >> 5;
  const int nWaves = (gridDim.x * blockDim.x) >> 5;

  float wsum = 0.0f;
  for (int t = waveId; t < ntgt; t += nWaves) {
    int s = (int)floorf(slice[t] * 10.0f);
    s = s < 0 ? 0 : (s > NSEC - 1 ? NSEC - 1 : s);
    const float* te = tgt + (long long)t * DIMS + lane * 8;
    const float* ce = centers + s * DIMS + lane * 8;
    const float4 t0 = *(const float4*)(te);
    const float4 t1 = *(const float4*)(te + 4);
    const float4 c0 = *(const float4*)(ce);
    const float4 c1 = *(const float4*)(ce + 4);
    float v = 0.0f, d;
    d = t0.x - c0.x + 1e-6f; v += d * d;
    d = t0.y - c0.y + 1e-6f; v += d * d;
    d = t0.z - c0.z + 1e-6f; v += d * d;
    d = t0.w - c0.w + 1e-6f; v += d * d;
    d = t1.x - c1.x + 1e-6f; v += d * d;
    d = t1.y - c1.y + 1e-6f; v += d * d;
    d = t1.z - c1.z + 1e-6f; v += d * d;
    d = t1.w - c1.w + 1e-6f; v += d * d;
#pragma unroll
    for (int off = 16; off > 0; off >>= 1) v += __shfl_xor(v, off);
    if (lane == 0) wsum += sqrtf(v);
  }
  if (lane == 0 && wsum != 0.0f) atomicAdd(acc, wsum);
}

__global__ void finalize_kernel(const float* __restrict__ acc,
                                float* __restrict__ out, float invN) {
  if (blockIdx.x == 0 && threadIdx.x == 0) out[0] = acc[0] * invN;
}

extern "C" void kernel_launch(void* const* d_in, const int* in_sizes, int n_in,
                              void* d_out, int out_size, void* d_ws, size_t ws_size,
                              hipStream_t stream) {
  const float* src_emb = (const float*)d_in[0];
  const int*   src_sec = (const int*)d_in[1];
  const float* tgt_emb = (const float*)d_in[2];
  const float* tgt_sl  = (const float*)d_in[3];
  const int nsrc = in_sizes[1];
  const int ntgt = in_sizes[3];

  float* ws     = (float*)d_ws;
  float* sums   = ws;                       // 16*256
  float* counts = ws + SEC_PAD * DIMS;      // 16
  float* acc    = counts + SEC_PAD;         // 1
  const int wsN = SEC_PAD * DIMS + SEC_PAD + 1;

  zero_ws_kernel<<<(wsN + 255) / 256, 256, 0, stream>>>(ws, wsN);

  const int nblk = (nsrc + CHUNK - 1) / CHUNK;
  segsum_wmma_kernel<<<nblk, 512, 0, stream>>>(src_emb, src_sec, sums, nsrc);
  count_kernel<<<256, 256, 0, stream>>>(src_sec, counts, nsrc);
  dist_kernel<<<1024, 256, 0, stream>>>(tgt_emb, tgt_sl, sums, counts, acc, ntgt);
  finalize_kernel<<<1, 32, 0, stream>>>(acc, (float*)d_out, 1.0f / (float)ntgt);
}